// LR_GCCF_80350248174014
// MI455X (gfx1250) — compile-verified
//
#include <hip/hip_runtime.h>
#include <hip/hip_bf16.h>

#define N_USERS 100000
#define N_ITEMS 50000
#define NF      64
#define N_EDGES 3200000
#define BATCH   16384
#define CHUNK   1024   // edges staged per block via TDM (3.2M / 1024 = 3125 exactly)

typedef float        v2f __attribute__((ext_vector_type(2)));
typedef float        v8f __attribute__((ext_vector_type(8)));
typedef unsigned int v4u __attribute__((ext_vector_type(4)));
typedef int          v4i __attribute__((ext_vector_type(4)));
typedef int          v8i __attribute__((ext_vector_type(8)));

// ---------------------------------------------------------------------------
// LDS byte-offset of a __shared__ object (generic AS pointer low 32 bits map
// to the group-segment offset on amdgcn).
__device__ __forceinline__ unsigned lds_off_of(const void* p) {
  return (unsigned)(unsigned long long)(uintptr_t)p;
}

// ---------------------------------------------------------------------------
// Tensor Data Mover: 1-D contiguous copy of `ndw` dwords global -> LDS.
// D# layout per CDNA5 ISA:
//   group0: [1:0]=count=1, [63:32]=lds_addr, [120:64]=global_addr, [127:126]=type=2
//   group1: data_size=4B, tensor_dim0=ndw, tensor_dim1=1, tile_dim0=ndw,
//           tensor_dim0_stride=ndw, everything else 0 (no pad/multicast/iterate)
__device__ __forceinline__ void tdm_load_1d(unsigned lds_off, const void* gsrc,
                                            unsigned ndw) {
  unsigned long long ga = (unsigned long long)(uintptr_t)gsrc;
  v4u g0;
  g0[0] = 1u;                                            // count=1, user D#
  g0[1] = lds_off;                                       // lds_addr
  g0[2] = (unsigned)ga;                                  // global_addr[31:0]
  g0[3] = (unsigned)((ga >> 32) & 0x01FFFFFFu) | (2u << 30); // addr[56:32] | type=2
  v8i g1;
  g1[0] = (int)(2u << 16);                               // data_size = 4 bytes
  g1[1] = (int)((ndw & 0xFFFFu) << 16);                  // tensor_dim0[15:0]
  g1[2] = (int)(((ndw >> 16) & 0xFFFFu) | (1u << 16));   // tensor_dim0[31:16], tensor_dim1=1
  g1[3] = (int)((ndw & 0xFFFFu) << 16);                  // tile_dim0 = ndw
  g1[4] = 0;                                             // tile_dim1/2 unused
  g1[5] = (int)ndw;                                      // tensor_dim0_stride[31:0]
  g1[6] = 0;
  g1[7] = 0;
  v4i z4 = {0, 0, 0, 0};
#if defined(__clang_major__) && (__clang_major__ >= 23)
  v8i z8 = {0, 0, 0, 0, 0, 0, 0, 0};
  __builtin_amdgcn_tensor_load_to_lds(g0, g1, z4, z4, z8, 0);
#else
  __builtin_amdgcn_tensor_load_to_lds(g0, g1, z4, z4, 0);
#endif
}

// ---------------------------------------------------------------------------
// dst[r][c] = src[r][c] * d[r]  via V_WMMA_F32_16X16X4_F32.
// One wave owns a 16x16 tile; D = sum_{c=0..3} A_c(16x4) * B_c(4x16), where
// A_c is the diagonal slice (cols 4c..4c+3 of diag(d)) and B_c holds rows
// 4c..4c+3 of the tile. ISA lane layouts (CDNA5 ISA §7.12.2):
//   A 16x4 : lanes 0-15 M=lane {K=0,1}, lanes 16-31 M=lane-16 {K=2,3}
//   B 4x16 : mirrored (N across lanes, K split the same way)
//   C/D    : vgpr j -> row j (lanes 0-15) / row 8+j (lanes 16-31), col=lane&15
__global__ __launch_bounds__(256) void scale_diag_wmma(
    const float* __restrict__ src, const float* __restrict__ d,
    float* __restrict__ dst, int nrows) {
  const int lane  = threadIdx.x & 31;
  const int wave  = threadIdx.x >> 5;
  const int tile  = blockIdx.x * 8 + wave;
  const int nTile = (nrows / 16) * (NF / 16);
  if (tile >= nTile) return;
  const int row0  = (tile >> 2) * 16;          // NF/16 == 4 column tiles
  const int col0  = (tile & 3) * 16;
  const int m     = lane & 15;
  const int khalf = lane >> 4;                 // 0: K pair {0,1}, 1: {2,3}
  const float dv  = d[row0 + m];

  v8f acc = {};
#pragma unroll
  for (int c = 0; c < 4; ++c) {
    const int k0 = 4 * c + 2 * khalf;
    v2f a, b;
    a.x = (m == k0)     ? dv : 0.0f;
    a.y = (m == k0 + 1) ? dv : 0.0f;
    b.x = src[(size_t)(row0 + k0)     * NF + col0 + m];
    b.y = src[(size_t)(row0 + k0 + 1) * NF + col0 + m];
    acc = __builtin_amdgcn_wmma_f32_16x16x4_f32(false, a, false, b,
                                                (short)0, acc, false, false);
  }
  const int rbase = (lane < 16) ? 0 : 8;
#pragma unroll
  for (int j = 0; j < 8; ++j)
    dst[(size_t)(row0 + rbase + j) * NF + (col0 + m)] = acc[j];
}

// ---------------------------------------------------------------------------
// One block stages CHUNK edges into LDS via TDM, then 64-lane groups process
// edges: coalesced 256B row gather from the previous layer, weighted
// global_atomic_add_f32 scatter into the new layer (both directions share the
// staged indices).
__global__ __launch_bounds__(256) void edge_scatter(
    const int* __restrict__ edge_user, const int* __restrict__ edge_item,
    const float* __restrict__ w_ui, const float* __restrict__ w_iu,
    const float* __restrict__ u_prev, const float* __restrict__ i_prev,
    float* __restrict__ u_new, float* __restrict__ i_new, int nEdges) {
  __shared__ int   s_u[CHUNK];
  __shared__ int   s_i[CHUNK];
  __shared__ float s_wui[CHUNK];
  __shared__ float s_wiu[CHUNK];

  const int base = blockIdx.x * CHUNK;
  const int cnt  = min(CHUNK, nEdges - base);

  if (threadIdx.x < 32) {                      // wave 0 issues the DMA
    tdm_load_1d(lds_off_of(s_u),   edge_user + base, (unsigned)cnt);
    tdm_load_1d(lds_off_of(s_i),   edge_item + base, (unsigned)cnt);
    tdm_load_1d(lds_off_of(s_wui), w_ui + base,      (unsigned)cnt);
    tdm_load_1d(lds_off_of(s_wiu), w_iu + base,      (unsigned)cnt);
    __builtin_amdgcn_s_wait_tensorcnt(0);
  }
  __syncthreads();

  const int grp = threadIdx.x >> 6;            // 4 edge-groups of 64 lanes
  const int f   = threadIdx.x & 63;
  for (int e = grp; e < cnt; e += 4) {
    const int   u  = s_u[e];
    const int   it = s_i[e];
    const float a  = s_wui[e];
    const float b  = s_wiu[e];
    const float iv = i_prev[(size_t)it * NF + f];
    const float uv = u_prev[(size_t)u  * NF + f];
    unsafeAtomicAdd(&u_new[(size_t)u  * NF + f], a * iv);
    unsafeAtomicAdd(&i_new[(size_t)it * NF + f], b * uv);
  }
}

// ---------------------------------------------------------------------------
// out rows: [users | pos_item | neg_item] x 256 ( = concat of 4 layer tables).
__global__ __launch_bounds__(256) void gather_out(
    const float* __restrict__ u0, const float* __restrict__ u1,
    const float* __restrict__ u2, const float* __restrict__ u3,
    const float* __restrict__ i0, const float* __restrict__ i1,
    const float* __restrict__ i2, const float* __restrict__ i3,
    const int* __restrict__ users, const int* __restrict__ pos,
    const int* __restrict__ neg, float* __restrict__ out) {
  const int row = blockIdx.x;                  // 0 .. 3*BATCH-1
  const int sec = row / BATCH;
  const int b   = row - sec * BATCH;
  int idx;
  const float *t0, *t1, *t2, *t3;
  if (sec == 0) { idx = users[b]; t0 = u0; t1 = u1; t2 = u2; t3 = u3; }
  else {
    idx = (sec == 1) ? pos[b] : neg[b];
    t0 = i0; t1 = i1; t2 = i2; t3 = i3;
  }
  const int t = threadIdx.x;
  const int k = t >> 6, f = t & 63;            // k uniform per wave
  const float* tbl = (k == 0) ? t0 : (k == 1) ? t1 : (k == 2) ? t2 : t3;
  out[(size_t)row * 256 + t] = tbl[(size_t)idx * NF + f];
}

// ---------------------------------------------------------------------------
extern "C" void kernel_launch(void* const* d_in, const int* in_sizes, int n_in,
                              void* d_out, int out_size, void* d_ws,
                              size_t ws_size, hipStream_t stream) {
  const int*   edge_user = (const int*)d_in[0];
  const int*   edge_item = (const int*)d_in[1];
  const float* ev_ui     = (const float*)d_in[2];
  const float* ev_iu     = (const float*)d_in[3];
  const float* d_users   = (const float*)d_in[4];
  const float* d_items   = (const float*)d_in[5];
  const float* emb_user  = (const float*)d_in[6];
  const float* emb_item  = (const float*)d_in[7];
  const int*   users     = (const int*)d_in[8];
  const int*   pos       = (const int*)d_in[9];
  const int*   neg       = (const int*)d_in[10];

  const size_t UT = (size_t)N_USERS * NF;
  const size_t IT = (size_t)N_ITEMS * NF;
  float* ws = (float*)d_ws;
  float* u[4];
  float* i[4];
  u[0] = (float*)emb_user;  // level-0 tables are read-only inputs
  i[0] = (float*)emb_item;
  u[1] = ws;            u[2] = ws + UT;   u[3] = ws + 2 * UT;
  i[1] = ws + 3 * UT;   i[2] = i[1] + IT; i[3] = i[2] + IT;

  const dim3 blk(256);
  const int uTiles = (N_USERS / 16) * (NF / 16);
  const int iTiles = (N_ITEMS / 16) * (NF / 16);
  const int eBlocks = (N_EDGES + CHUNK - 1) / CHUNK;

  for (int L = 1; L <= 3; ++L) {
    scale_diag_wmma<<<(uTiles + 7) / 8, blk, 0, stream>>>(u[L - 1], d_users,
                                                          u[L], N_USERS);
    scale_diag_wmma<<<(iTiles + 7) / 8, blk, 0, stream>>>(i[L - 1], d_items,
                                                          i[L], N_ITEMS);
    edge_scatter<<<eBlocks, blk, 0, stream>>>(edge_user, edge_item, ev_ui,
                                              ev_iu, u[L - 1], i[L - 1],
                                              u[L], i[L], N_EDGES);
  }
  gather_out<<<3 * BATCH, blk, 0, stream>>>(u[0], u[1], u[2], u[3],
                                            i[0], i[1], i[2], i[3],
                                            users, pos, neg, (float*)d_out);
}